// RNNModel_39256001086067
// MI455X (gfx1250) — compile-verified
//
#include <hip/hip_runtime.h>

// ---------------------------------------------------------------------------
// Elman ReLU RNN, fp32, MI455X (gfx1250).
//   B=256, T=2048, I=50, H=116, O=50
// 3 kernels:
//   k1: xp[b,t,h] = x[b,t,:]@W_ih[h,:] + b_ih[h] + b_hh[h]   (parallel, WMMA f32)
//   k2: h_t = relu(xp_t + h_{t-1}@W_hh^T), hs overwrites xp  (sequential over T)
//   k3: out[b,t,o] = hs[b,t,:]@W_fc[o,:] + b_fc[o]           (parallel, WMMA f32)
// ---------------------------------------------------------------------------

typedef float v2f __attribute__((ext_vector_type(2)));
typedef float v8f __attribute__((ext_vector_type(8)));

#define B_ 256
#define T_ 2048
#define I_ 50
#define H_ 116
#define O_ 50

// D = A(16x4 f32) * B(4x16 f32) + C(16x16 f32)
__device__ __forceinline__ v8f wmma4(v2f a, v2f b, v8f c) {
  return __builtin_amdgcn_wmma_f32_16x16x4_f32(
      /*neg_a=*/false, a, /*neg_b=*/false, b,
      /*c_mod=*/(short)0, c, /*reuse_a=*/false, /*reuse_b=*/false);
}

// Per-lane fragment coordinates (wave32, 16x16x4 f32 layout):
//   A: row m = lane&15, k = 4*ks + 2*(lane>>4) .. +1   (2 contiguous floats of row m)
//   B: col n = lane&15, k = 4*ks + 2*(lane>>4) .. +1   (2 contiguous floats of W row n)
//   C/D: vgpr r <-> (M = r + 8*(lane>>4), N = lane&15)

// ---------------------------------------------------------------------------
// Kernel 1: xp = x @ W_ih^T + (b_ih + b_hh).  Rows = B*T, K = 50 (13 ksteps,
// cols 50..51 zero-padded), N = 116 padded to 128 (8 waves x 16 cols).
// 128 rows per block -> 4096 blocks, memory-bound streaming of x.
// ---------------------------------------------------------------------------
__global__ __launch_bounds__(256) void rnn_xproj(const float* __restrict__ x,
                                                 const float* __restrict__ Wih,
                                                 const float* __restrict__ bih,
                                                 const float* __restrict__ bhh,
                                                 float* __restrict__ xp) {
  __shared__ float xs[128 * 54];   // stride 54: 54*m mod 64 distinct for m=0..15
  __shared__ float ws[128 * 54];
  const int tid   = threadIdx.x;
  const int lane  = tid & 31;
  const int wave  = tid >> 5;
  const int nl    = lane & 15;
  const int khalf = (lane >> 4) * 2;
  const int mrow  = (lane >> 4) * 8;
  const int rb0   = blockIdx.x * 128;

  for (int idx = tid; idx < 128 * 52; idx += 256) {
    const int a = idx / 52, kk = idx % 52;
    ws[a * 54 + kk] = (a < H_ && kk < I_) ? Wih[a * I_ + kk] : 0.0f;
    xs[a * 54 + kk] = (kk < I_) ? x[(size_t)(rb0 + a) * I_ + kk] : 0.0f;
  }
  __syncthreads();

  const int n = wave * 16 + nl;                       // output H column
  const float biasv = (n < H_) ? (bih[n] + bhh[n]) : 0.0f;

  for (int mt = 0; mt < 8; ++mt) {
    v8f acc;
#pragma unroll
    for (int r = 0; r < 8; ++r) acc[r] = biasv;
#pragma unroll
    for (int ks = 0; ks < 13; ++ks) {
      v2f a = *(const v2f*)&xs[(mt * 16 + nl) * 54 + ks * 4 + khalf];
      v2f b = *(const v2f*)&ws[n * 54 + ks * 4 + khalf];
      acc = wmma4(a, b, acc);
    }
    if (n < H_) {
#pragma unroll
      for (int r = 0; r < 8; ++r)
        xp[(size_t)(rb0 + mt * 16 + r + mrow) * H_ + n] = acc[r];
    }
  }
}

// ---------------------------------------------------------------------------
// Kernel 2: the sequential recurrence.  16 blocks x 16 batch rows; 8 waves,
// wave w owns H-columns [16w, 16w+16).  W_hh fragments live in 58 VGPRs
// (loaded once); h_{t-1} lives in a double-buffered LDS tile (stride 132 ->
// conflict-free ds_load_b64).  29 WMMAs/step split into 2 chains.  One
// s_barrier per step.  h_t overwrites xp in place (becomes hs for kernel 3).
// ---------------------------------------------------------------------------
__global__ __launch_bounds__(256) void rnn_recur(const float* __restrict__ Whh,
                                                 float* __restrict__ xp) {
  __shared__ float hbuf[2][16 * 132];
  const int tid   = threadIdx.x;
  const int lane  = tid & 31;
  const int wave  = tid >> 5;
  const int nl    = lane & 15;
  const int khalf = (lane >> 4) * 2;
  const int mrow  = (lane >> 4) * 8;
  const int row0  = blockIdx.x * 16;                  // batch rows
  const int n     = wave * 16 + nl;                   // H column

  // Preload B fragments of W_hh^T: B[k][n] = W_hh[n][k] -> row-major b64 loads.
  v2f bfrag[29];
#pragma unroll
  for (int ks = 0; ks < 29; ++ks) {
    if (n < H_) bfrag[ks] = *(const v2f*)(Whh + (size_t)n * H_ + ks * 4 + khalf);
    else        { bfrag[ks].x = 0.0f; bfrag[ks].y = 0.0f; }
  }

  for (int i = tid; i < 2 * 16 * 132; i += 256) (&hbuf[0][0])[i] = 0.0f;  // h_{-1}=0
  __syncthreads();

  size_t rowptr[8];                                   // xp offsets of this lane's C rows
#pragma unroll
  for (int r = 0; r < 8; ++r)
    rowptr[r] = (size_t)(row0 + r + mrow) * T_ * H_ + n;

  for (int t = 0; t < T_; ++t) {
    const int cur = t & 1, nxt = cur ^ 1;
    const size_t toff = (size_t)t * H_;

    v8f acc0, acc1;                                   // C init = xp_t (has biases)
#pragma unroll
    for (int r = 0; r < 8; ++r) {
      float c = 0.0f;
      if (n < H_) c = xp[rowptr[r] + toff];
      acc0[r] = c;
      acc1[r] = 0.0f;
    }
    if (t + 1 < T_) {                                 // speculative prefetch of xp_{t+1}
#pragma unroll
      for (int r = 0; r < 8; ++r)
        __builtin_prefetch(&xp[rowptr[r] + toff + H_], 0, 0);
    }

    const float* hrow = &hbuf[cur][nl * 132];         // A row for this lane
#pragma unroll
    for (int ks = 0; ks < 29; ++ks) {
      v2f a = *(const v2f*)(hrow + ks * 4 + khalf);
      if (ks & 1) acc1 = wmma4(a, bfrag[ks], acc1);
      else        acc0 = wmma4(a, bfrag[ks], acc0);
    }

#pragma unroll
    for (int r = 0; r < 8; ++r) {
      float h = fmaxf(acc0[r] + acc1[r], 0.0f);       // ReLU cell
      hbuf[nxt][(r + mrow) * 132 + n] = h;            // publish h_t for next step
      if (n < H_) xp[rowptr[r] + toff] = h;           // hs overwrites xp in place
    }
    __syncthreads();
  }
}

// ---------------------------------------------------------------------------
// Kernel 3: out = hs @ W_fc^T + b_fc.  Rows = B*T, K = 116 (29 ksteps),
// N = 50 padded to 64 (4 n-tiles; 8 waves = 4 n-tiles x 2 m-halves).
// 128 rows per block -> 4096 blocks, memory-bound streaming of hs.
// ---------------------------------------------------------------------------
__global__ __launch_bounds__(256) void rnn_fc(const float* __restrict__ hs,
                                              const float* __restrict__ Wfc,
                                              const float* __restrict__ bfc,
                                              float* __restrict__ out) {
  __shared__ float ht[128 * 132];
  __shared__ float wt[64 * 118];                      // 118 mod 64 = 54: conflict-free
  const int tid   = threadIdx.x;
  const int lane  = tid & 31;
  const int wave  = tid >> 5;
  const int nl    = lane & 15;
  const int khalf = (lane >> 4) * 2;
  const int mrow  = (lane >> 4) * 8;
  const int rb0   = blockIdx.x * 128;

  for (int idx = tid; idx < 128 * H_; idx += 256) {
    const int rr = idx / H_, kk = idx % H_;
    ht[rr * 132 + kk] = hs[(size_t)(rb0 + rr) * H_ + kk];
  }
  for (int idx = tid; idx < 64 * H_; idx += 256) {
    const int nn = idx / H_, kk = idx % H_;
    wt[nn * 118 + kk] = (nn < O_) ? Wfc[nn * H_ + kk] : 0.0f;
  }
  __syncthreads();

  const int n   = (wave & 3) * 16 + nl;               // output O column
  const int mt0 = (wave >> 2) * 4;                    // this wave's 4 m-tiles
  const float bv = (n < O_) ? bfc[n] : 0.0f;

  for (int mt = mt0; mt < mt0 + 4; ++mt) {
    v8f acc0, acc1;
#pragma unroll
    for (int r = 0; r < 8; ++r) { acc0[r] = bv; acc1[r] = 0.0f; }
#pragma unroll
    for (int ks = 0; ks < 29; ++ks) {
      v2f a = *(const v2f*)&ht[(mt * 16 + nl) * 132 + ks * 4 + khalf];
      v2f b = *(const v2f*)&wt[n * 118 + ks * 4 + khalf];
      if (ks & 1) acc1 = wmma4(a, b, acc1);
      else        acc0 = wmma4(a, b, acc0);
    }
    if (n < O_) {
#pragma unroll
      for (int r = 0; r < 8; ++r)
        out[(size_t)(rb0 + mt * 16 + r + mrow) * O_ + n] = acc0[r] + acc1[r];
    }
  }
}

// ---------------------------------------------------------------------------
extern "C" void kernel_launch(void* const* d_in, const int* in_sizes, int n_in,
                              void* d_out, int out_size, void* d_ws, size_t ws_size,
                              hipStream_t stream) {
  (void)in_sizes; (void)n_in; (void)out_size; (void)ws_size;
  const float* x   = (const float*)d_in[0];
  const float* Wih = (const float*)d_in[1];
  const float* bih = (const float*)d_in[2];
  const float* Whh = (const float*)d_in[3];
  const float* bhh = (const float*)d_in[4];
  const float* Wfc = (const float*)d_in[5];
  const float* bfc = (const float*)d_in[6];
  float* out = (float*)d_out;
  float* xp  = (float*)d_ws;                          // B*T*H floats (xp -> hs in place)

  rnn_xproj<<<(B_ * T_) / 128, 256, 0, stream>>>(x, Wih, bih, bhh, xp);
  rnn_recur<<<B_ / 16, 256, 0, stream>>>(Whh, xp);
  rnn_fc<<<(B_ * T_) / 128, 256, 0, stream>>>(xp, Wfc, bfc, out);
}